// SchNetInit_3874060501585
// MI455X (gfx1250) — compile-verified
//
#include <hip/hip_runtime.h>
#include <hip/hip_bf16.h>
#include <stdint.h>

// ---------------------------------------------------------------------------
// SchNet init on MI455X (gfx1250): pure store-bandwidth problem (~704 MB out,
// ~30us floor at 23.3 TB/s). Strategy:
//  * node_embs : async-stage the 51KB table into LDS (CDNA5
//                global_load_async_to_lds_b128 + s_wait_asynccnt), then one
//                wave32 per node: lane l does ds_load_b128 of float4 l of the
//                row (bank-conflict-free) and a non-temporal b128 store.
//  * edge_embs : 256-edge tile per block. Phase A: one thread per edge
//                computes the distance (coords are L2-resident, 1.2MB),
//                NT-stores edge_weights, parks d in LDS. Phase B: block
//                writes the 256x50 tile as flat coalesced NT float4 stores.
// ---------------------------------------------------------------------------

typedef float v4f __attribute__((ext_vector_type(4)));

#define H_NF 128
#define H_V4 (H_NF / 4)          // 32 float4 per embedding row
#define POSS_ELEMS 100
#define TABLE_F4 (POSS_ELEMS * H_V4)  // 3200 float4 = 51200 B of LDS
#define N_GAUSS 50
#define CUTOFF_F 10.0f
#define TILE 256

// ------------------------- node embedding gather ---------------------------
__global__ void __launch_bounds__(256)
schnet_node_emb_kernel(const int* __restrict__ atomic_ns,
                       const float* __restrict__ weight,
                       float* __restrict__ out,
                       int n_nodes, int table_f4_valid) {
    __shared__ v4f table[TABLE_F4];

    // --- stage the whole embedding table into LDS with async tensor-path ---
    unsigned lds_base = (unsigned)(uintptr_t)(&table[0]);   // LDS offset (low 32b)
    const v4f* w4 = (const v4f*)weight;
    for (int c = threadIdx.x; c < table_f4_valid; c += blockDim.x) {
        unsigned loff = lds_base + (unsigned)c * 16u;
        const v4f* gptr = w4 + c;
        asm volatile("global_load_async_to_lds_b128 %0, %1, off"
                     :: "v"(loff), "v"(gptr)
                     : "memory");
    }
    asm volatile("s_wait_asynccnt 0" ::: "memory");
    __syncthreads();

    // --- serve gathers from LDS: one wave32 per node ---
    const int lane            = threadIdx.x & 31;
    const int wave_in_block   = threadIdx.x >> 5;
    const int waves_per_block = blockDim.x >> 5;
    const int gwave  = blockIdx.x * waves_per_block + wave_in_block;
    const int nwaves = gridDim.x * waves_per_block;

    v4f* out4 = (v4f*)out;
    for (int n = gwave; n < n_nodes; n += nwaves) {
        int row = atomic_ns[n];                       // broadcast within wave
        v4f v = table[row * H_V4 + lane];             // ds_load_b128, no conflicts
        __builtin_nontemporal_store(v, out4 + (size_t)n * H_V4 + lane);
    }
}

// ----------------- edge distances + gaussian smearing ----------------------
__global__ void __launch_bounds__(TILE)
schnet_edge_kernel(const int* __restrict__ edge_index,
                   const float* __restrict__ coords,
                   float* __restrict__ edge_embs,
                   float* __restrict__ edge_w,
                   int n_edges) {
    __shared__ float sdist[TILE];

    const int   t    = threadIdx.x;
    const size_t base = (size_t)blockIdx.x * TILE;
    const int   e    = (int)base + t;

    // Phase A: distance per edge
    float d = 0.0f;
    if (e < n_edges) {
        int i = edge_index[e];            // edge_index[0][e]
        int j = edge_index[n_edges + e];  // edge_index[1][e]
        float dx = coords[3 * i + 0] - coords[3 * j + 0];
        float dy = coords[3 * i + 1] - coords[3 * j + 1];
        float dz = coords[3 * i + 2] - coords[3 * j + 2];
        d = sqrtf(dx * dx + dy * dy + dz * dz);
        __builtin_nontemporal_store(d, edge_w + e);
    }
    sdist[t] = d;
    __syncthreads();

    // Phase B: gaussian smearing, flat coalesced tile writes
    const float delta = CUTOFF_F / (float)(N_GAUSS - 1);
    const float coeff = -0.5f / (delta * delta);
    float* out_base = edge_embs + base * N_GAUSS;

    int remaining = n_edges - (int)base;
    int valid = remaining < TILE ? remaining : TILE;

    if (valid == TILE) {
        // full tile: 256*50 floats = 3200 float4, all in-bounds, b128 NT stores
        v4f* o4 = (v4f*)out_base;
        const int total4 = TILE * N_GAUSS / 4;   // 3200
        for (int c = t; c < total4; c += TILE) {
            int f = c * 4;
            v4f val;
#pragma unroll
            for (int k = 0; k < 4; ++k) {
                int idx = f + k;
                int el  = idx / N_GAUSS;          // const-div -> mulhi
                int g   = idx - el * N_GAUSS;
                float x = sdist[el] - (float)g * delta;
                val[k] = __expf(coeff * x * x);   // v_exp_f32
            }
            __builtin_nontemporal_store(val, o4 + c);
        }
    } else if (valid > 0) {
        // ragged tail tile: scalar guarded NT stores
        int totalf = valid * N_GAUSS;
        for (int c = t; c < totalf; c += TILE) {
            int el = c / N_GAUSS;
            int g  = c - el * N_GAUSS;
            float x = sdist[el] - (float)g * delta;
            __builtin_nontemporal_store(__expf(coeff * x * x), out_base + c);
        }
    }
}

// ---------------------------------------------------------------------------
extern "C" void kernel_launch(void* const* d_in, const int* in_sizes, int n_in,
                              void* d_out, int out_size, void* d_ws, size_t ws_size,
                              hipStream_t stream) {
    (void)n_in; (void)out_size; (void)d_ws; (void)ws_size;

    const int*   atomic_ns  = (const int*)d_in[0];
    const int*   edge_index = (const int*)d_in[1];
    const float* coords     = (const float*)d_in[2];
    // d_in[3] = batch_node_vec (unused by the reference computation)
    const float* weight     = (const float*)d_in[4];

    const int n_nodes  = in_sizes[0];
    const int n_edges  = in_sizes[1] / 2;
    const int table_f4 = in_sizes[4] / 4;   // #float4 in embedding table
    const int stage_f4 = table_f4 < TABLE_F4 ? table_f4 : TABLE_F4;

    // output layout: [node_embs N*128][edge_embs E*50][edge_weights E]
    float* node_embs = (float*)d_out;
    float* edge_embs = node_embs + (size_t)n_nodes * H_NF;
    float* edge_w    = edge_embs + (size_t)n_edges * N_GAUSS;

    // node kernel: grid-stride, 8 wave32/block, one node per wave per step
    int node_blocks = (n_nodes + 7) / 8;
    if (node_blocks > 1024) node_blocks = 1024;
    if (node_blocks < 1) node_blocks = 1;
    schnet_node_emb_kernel<<<node_blocks, 256, 0, stream>>>(
        atomic_ns, weight, node_embs, n_nodes, stage_f4);

    // edge kernel: one 256-edge tile per block
    int edge_blocks = (n_edges + TILE - 1) / TILE;
    if (edge_blocks < 1) edge_blocks = 1;
    schnet_edge_kernel<<<edge_blocks, TILE, 0, stream>>>(
        edge_index, coords, edge_embs, edge_w, n_edges);
}